// IterativeCorrelationLayer_12833362280729
// MI455X (gfx1250) — compile-verified
//
#include <hip/hip_runtime.h>
#include <hip/hip_bf16.h>
#include <math.h>

typedef __attribute__((ext_vector_type(2))) float v2f;
typedef __attribute__((ext_vector_type(8))) float v8f;

#define BATCH   8
#define CHANS   256
#define HEIGHT  64
#define WIDTH   128
#define GROUPS  4
#define CG      64          // channels per group
#define NSHIFT  9           // 1x9 window along x
#define WPAD    132         // padded LDS row stride (bank spread)

// One block per (b, h, g) row-group. 256 threads = 8 wave32.
//  Phase 0 : per-pixel bilinear corner offsets + weights (zero-pad folded in)
//  Phase 1 : warp the 64x128 group row of `right` into LDS (bilinear gather)
//  Phase 2 : per-wave banded Gram matrix via v_wmma_f32_16x16x4_f32,
//            band extracted through wave-private LDS staging, scaled by 1/64.
__global__ __launch_bounds__(256, 1)
void iter_corr_wmma_kernel(const float* __restrict__ left,
                           const float* __restrict__ right,
                           const float* __restrict__ flow,
                           float* __restrict__ out)
{
    __shared__ float s_warp[CG][WPAD];        // warped group row      (~33.8 KB)
    __shared__ int   s_off[4][WIDTH];         // corner plane offsets  (2 KB)
    __shared__ float s_wgt[4][WIDTH];         // corner weights        (2 KB)
    __shared__ float s_ct[8][2][16][16];      // per-wave D staging    (16 KB)

    const int blk = blockIdx.x;
    const int g   = blk & 3;
    const int h   = (blk >> 2) & (HEIGHT - 1);
    const int b   = blk >> 8;
    const int tid = threadIdx.x;

    // ---- Phase 0: sampling tables for this row (threads 0..127) ----
    if (tid < WIDTH) {
        const int w = tid;
        const float fx = flow[((size_t)(b * 2 + 0) * HEIGHT + h) * WIDTH + w];
        const float fy = flow[((size_t)(b * 2 + 1) * HEIGHT + h) * WIDTH + w];
        const float xs = (float)w + fx;
        const float ys = (float)h + fy;
        const float x0f = floorf(xs), y0f = floorf(ys);
        const float wx1 = xs - x0f, wx0 = 1.0f - wx1;
        const float wy1 = ys - y0f, wy0 = 1.0f - wy1;
        const int x0 = (int)x0f, y0 = (int)y0f;
#pragma unroll
        for (int j = 0; j < 4; ++j) {
            const int xi = x0 + (j & 1);
            const int yi = y0 + (j >> 1);
            const bool valid = (xi >= 0) && (xi < WIDTH) && (yi >= 0) && (yi < HEIGHT);
            const int xc = min(max(xi, 0), WIDTH - 1);
            const int yc = min(max(yi, 0), HEIGHT - 1);
            const float wgt = ((j & 1) ? wx1 : wx0) * ((j >> 1) ? wy1 : wy0);
            s_off[j][w] = yc * WIDTH + xc;
            s_wgt[j][w] = valid ? wgt : 0.0f;
        }
    }
    __syncthreads();

    // ---- Phase 1: bilinear-warp the group row of `right` into LDS ----
    {
        const float* gbase = right + (size_t)(b * CHANS + g * CG) * (HEIGHT * WIDTH);
        for (int idx = tid; idx < CG * WIDTH; idx += 256) {
            const int c = idx >> 7;           // 0..63
            const int w = idx & (WIDTH - 1);  // 0..127
            const float* plane = gbase + (size_t)c * (HEIGHT * WIDTH);
            float acc = s_wgt[0][w] * plane[s_off[0][w]];
            acc = fmaf(s_wgt[1][w], plane[s_off[1][w]], acc);
            acc = fmaf(s_wgt[2][w], plane[s_off[2][w]], acc);
            acc = fmaf(s_wgt[3][w], plane[s_off[3][w]], acc);
            s_warp[c][w] = acc;
        }
    }
    __syncthreads();

    // ---- Phase 2: banded Gram matrix with f32 WMMA ----
    const int wv   = tid >> 5;        // wave id 0..7 -> pixel tile
    const int lane = tid & 31;
    const int lmod = lane & 15;       // M (A) / N (B) slot
    const int lhi  = lane >> 4;       // K half selector
    const int w0   = wv * 16;

    // A tile (left): 16 pixels x 64 channels, registers only, ISA 16x4 layout:
    //   VGPR0 holds K=0 (lanes 0-15) / K=2 (lanes 16-31); VGPR1 holds K=1 / K=3.
    v2f A[16];
    {
        const float* lbase = left +
            ((size_t)(b * CHANS + g * CG) * HEIGHT + h) * WIDTH + (w0 + lmod);
#pragma unroll
        for (int k = 0; k < 16; ++k) {
            const int c = 4 * k + 2 * lhi;
            v2f a;
            a.x = lbase[(size_t)c * (HEIGHT * WIDTH)];
            a.y = lbase[(size_t)(c + 1) * (HEIGHT * WIDTH)];
            A[k] = a;
        }
    }

    // Two 16-wide N-tiles cover the clip(w+dx-4) band: bases w0-4 and w0+12.
#pragma unroll
    for (int nt = 0; nt < 2; ++nt) {
        const int basew = w0 - 4 + nt * 16;
        const int wcol  = min(max(basew + lmod, 0), WIDTH - 1); // edge clamp
        v8f acc = {};
#pragma unroll
        for (int k = 0; k < 16; ++k) {
            const int c = 4 * k + 2 * lhi;
            v2f Bv;
            Bv.x = s_warp[c][wcol];
            Bv.y = s_warp[c + 1][wcol];
            acc = __builtin_amdgcn_wmma_f32_16x16x4_f32(
                /*neg_a=*/false, A[k], /*neg_b=*/false, Bv,
                /*c_mod=*/(short)0, acc, /*reuse_a=*/false, /*reuse_b=*/false);
        }
        // D layout: VGPR r -> (M = r + 8*lhi, N = lmod); stage to LDS (wave-private)
#pragma unroll
        for (int r = 0; r < 8; ++r) {
            s_ct[wv][nt][r + 8 * lhi][lmod] = acc[r];
        }
    }

    // Band extraction: 16 pixels x 9 shifts = 144 outputs per wave.
    for (int e = lane; e < 16 * NSHIFT; e += 32) {
        const int dx = e >> 4;                       // 0..8
        const int m  = e & 15;                       // pixel in tile
        const int w  = w0 + m;
        const int wp = min(max(w + dx - 4, 0), WIDTH - 1);
        const int n  = wp - (w0 - 4);                // 0..23 across the two tiles
        const float val = s_ct[wv][n >> 4][m][n & 15] * (1.0f / (float)CG);
        out[(((size_t)b * (GROUPS * NSHIFT) + g * NSHIFT + dx) * HEIGHT + h) * WIDTH + w] = val;
    }
}

extern "C" void kernel_launch(void* const* d_in, const int* in_sizes, int n_in,
                              void* d_out, int out_size, void* d_ws, size_t ws_size,
                              hipStream_t stream)
{
    const float* left  = (const float*)d_in[0];
    const float* right = (const float*)d_in[1];
    const float* flow  = (const float*)d_in[2];
    float* out = (float*)d_out;

    const int nblocks = BATCH * HEIGHT * GROUPS;   // 2048 blocks of (b,h,g)
    iter_corr_wmma_kernel<<<nblocks, 256, 0, stream>>>(left, right, flow, out);
}